// NodeCell_20761871909637
// MI455X (gfx1250) — compile-verified
//
#include <hip/hip_runtime.h>

typedef __attribute__((ext_vector_type(2))) float v2f;
typedef __attribute__((ext_vector_type(8))) float v8f;

#define HDIM 16
#define NCOLS 224      // stage-A combined output cols: ifo(48) a(16) gates(128) n1(16) n2x(16)
#define NTOT  240      // + w_2h (16) for stage B
#define WB_S  36       // LDS stride (floats) for weight tiles [n][k], k=0..31 (even -> 8B aligned v2f)
#define XH_S  34       // LDS stride (floats) for [x|h] tile [m][k], k=0..31

struct KParams {
  const float *inputs, *h_prev, *c_prev;
  const float *w_in, *b_in;
  const float *w_ifo_x, *b_ifo_x, *w_ifo_h, *b_ifo_h;
  const float *w_a_x, *b_a_x, *w_a_h, *b_a_h;
  const float *w_cx, *b_cx, *w_ch, *b_ch;
  const float *w_1x, *b_1x, *w_1h, *b_1h;
  const float *w_2x, *b_2x, *w_2h, *b_2h;
  float *out_h, *out_c, *out_n;   // d_out sections
  float *gw, *n1w;                // workspace: gates [B,128], n1 [B,16]
  int nTiles;                     // B/16
};

// sigmoid with native-rate exp + rcp: ~4 VALU ops instead of IEEE division sequence
__device__ __forceinline__ float sigf(float x) {
  return __builtin_amdgcn_rcpf(1.0f + __expf(-x));
}
__device__ __forceinline__ float tanhfast(float x) {
  // tanh(x) = 2*sigmoid(2x) - 1 ; saturates gracefully to +/-1
  return fmaf(2.0f, sigf(2.0f * x), -1.0f);
}

// combined weight fetch: columns n in [0,240), K rows k in [0,32): k<16 -> x-side, k>=16 -> h-side
__device__ __forceinline__ float wfetch(const KParams& p, int n, int k) {
  if (n < 48)  { return (k < 16) ? p.w_ifo_x[n * 16 + k]        : p.w_ifo_h[n * 16 + k - 16]; }
  if (n < 64)  { int m = n - 48;  return (k < 16) ? p.w_a_x[m * 16 + k] : p.w_a_h[m * 16 + k - 16]; }
  if (n < 192) { int m = n - 64;  return (k < 16) ? p.w_cx[m * 16 + k]  : p.w_ch[m * 16 + k - 16]; }
  if (n < 208) { int m = n - 192; return (k < 16) ? p.w_1x[m * 16 + k]  : p.w_1h[m * 16 + k - 16]; }
  if (n < 224) { int m = n - 208; return (k < 16) ? p.w_2x[m * 16 + k]  : 0.0f; }  // zero h-half: n2x only sees x
  { int m = n - 224; return (k < 16) ? p.w_2h[m * 16 + k] : 0.0f; }
}
__device__ __forceinline__ float bfetch(const KParams& p, int n) {
  if (n < 48)  return p.b_ifo_x[n] + p.b_ifo_h[n];
  if (n < 64)  return p.b_a_x[n - 48] + p.b_a_h[n - 48];
  if (n < 192) return p.b_cx[n - 64] + p.b_ch[n - 64];
  if (n < 208) return p.b_1x[n - 192] + p.b_1h[n - 192];
  if (n < 224) return p.b_2x[n - 208] + p.b_2h[n - 208]; // full n2 bias added in stage A
  return 0.0f;
}

__launch_bounds__(256)
__global__ void nodecell_gemm_kernel(KParams p) {
  __shared__ float WB[NTOT * WB_S];
  __shared__ float BIAS[NCOLS];
  __shared__ float XH[8][16 * XH_S];   // per-wave [m=0..15][k=0..31] tile

  const int tid = threadIdx.x;

  // ---- one-time per block: stage all weights / combined biases into LDS ----
  for (int e = tid; e < NTOT * 32; e += 256) {
    int n = e >> 5, k = e & 31;
    WB[n * WB_S + k] = wfetch(p, n, k);
  }
  for (int n = tid; n < NCOLS; n += 256) BIAS[n] = bfetch(p, n);
  __syncthreads();

  const int lane = tid & 31;
  const int wave = tid >> 5;
  const int lo = lane & 15;        // N index within tile / A-matrix M row
  const int hi = lane >> 4;        // half-wave select
  float* xh = XH[wave];

  const float win = p.w_in[lo];
  const float bin = p.b_in[lo];

  const int waveGlobal = blockIdx.x * 8 + wave;
  const int nWaves = gridDim.x * 8;

  for (int T = waveGlobal; T < p.nTiles; T += nWaves) {
    const int mbase = T * 16;

    // ---- stage x = relu(inputs*w_in + b_in) (k 0..15) and h_prev (k 16..31) into LDS [m][k] ----
    #pragma unroll
    for (int t = 0; t < 8; ++t) {
      int m = 2 * t + hi;
      float xv = fmaf(p.inputs[mbase + m], win, bin);
      xv = xv > 0.0f ? xv : 0.0f;
      xh[m * XH_S + lo] = xv;
      // coalesced h_prev load: flat = 32t + lane -> (m = 2t+hi, k = lo)
      xh[m * XH_S + 16 + lo] = p.h_prev[(size_t)mbase * HDIM + t * 32 + lane];
    }

    v8f hv;  // new h, kept in registers for stage B / n output

    // ================= PASS 1: tiles 0..6  (i,f,o,a, gates cols 0..47) =================
    v8f acc[7];
    #pragma unroll
    for (int t = 0; t < 7; ++t) {
      float b = BIAS[t * 16 + lo];
      acc[t] = (v8f){b, b, b, b, b, b, b, b};
    }
    #pragma unroll
    for (int ks = 0; ks < 8; ++ks) {
      int k0 = ks * 4 + 2 * hi;
      v2f A = *reinterpret_cast<const v2f*>(&xh[lo * XH_S + k0]);
      #pragma unroll
      for (int t = 0; t < 7; ++t) {
        v2f Bf = *reinterpret_cast<const v2f*>(&WB[(t * 16 + lo) * WB_S + k0]);
        acc[t] = __builtin_amdgcn_wmma_f32_16x16x4_f32(false, A, false, Bf,
                                                       (short)0, acc[t], false, false);
      }
    }
    #pragma unroll
    for (int r = 0; r < 8; ++r) {
      size_t m = (size_t)(mbase + r + 8 * hi);
      float iv = sigf(acc[0][r]);
      float fv = sigf(acc[1][r]);
      float ov = sigf(acc[2][r]);
      float av = tanhfast(acc[3][r]);
      float cp = p.c_prev[m * HDIM + lo];
      float cv = fmaf(iv, av, fv * cp);
      p.out_c[m * HDIM + lo] = cv;
      float hvv = ov * tanhfast(cv);
      p.out_h[m * HDIM + lo] = hvv;
      hv[r] = hvv;
      p.gw[m * 128 +  0 + lo] = sigf(acc[4][r]);
      p.gw[m * 128 + 16 + lo] = sigf(acc[5][r]);
      p.gw[m * 128 + 32 + lo] = sigf(acc[6][r]);
    }

    // ================= PASS 2: tiles 7..13 (gates cols 48..127, n1, n2x) =================
    v8f acc2[7];
    #pragma unroll
    for (int t = 0; t < 7; ++t) {
      float b = BIAS[(t + 7) * 16 + lo];
      acc2[t] = (v8f){b, b, b, b, b, b, b, b};
    }
    #pragma unroll
    for (int ks = 0; ks < 8; ++ks) {
      int k0 = ks * 4 + 2 * hi;
      v2f A = *reinterpret_cast<const v2f*>(&xh[lo * XH_S + k0]);
      #pragma unroll
      for (int t = 0; t < 7; ++t) {
        v2f Bf = *reinterpret_cast<const v2f*>(&WB[((t + 7) * 16 + lo) * WB_S + k0]);
        acc2[t] = __builtin_amdgcn_wmma_f32_16x16x4_f32(false, A, false, Bf,
                                                        (short)0, acc2[t], false, false);
      }
    }
    #pragma unroll
    for (int r = 0; r < 8; ++r) {
      size_t m = (size_t)(mbase + r + 8 * hi);
      p.gw[m * 128 +  48 + lo] = sigf(acc2[0][r]);
      p.gw[m * 128 +  64 + lo] = sigf(acc2[1][r]);
      p.gw[m * 128 +  80 + lo] = sigf(acc2[2][r]);
      p.gw[m * 128 +  96 + lo] = sigf(acc2[3][r]);
      p.gw[m * 128 + 112 + lo] = sigf(acc2[4][r]);
      p.n1w[m * HDIM + lo]     = sigf(acc2[5][r]);
    }

    // ================= STAGE B: n2 = sigmoid(n2x + h @ w_2h^T + bias) =================
    // restage new h into the k=0..15 region of xh (x no longer needed)
    #pragma unroll
    for (int r = 0; r < 8; ++r) {
      xh[(r + 8 * hi) * XH_S + lo] = hv[r];
    }
    v8f a3 = acc2[6];  // n2x + full n2 bias
    #pragma unroll
    for (int ks = 0; ks < 4; ++ks) {
      int k0 = ks * 4 + 2 * hi;
      v2f A  = *reinterpret_cast<const v2f*>(&xh[lo * XH_S + k0]);
      v2f Bf = *reinterpret_cast<const v2f*>(&WB[(224 + lo) * WB_S + k0]);
      a3 = __builtin_amdgcn_wmma_f32_16x16x4_f32(false, A, false, Bf,
                                                 (short)0, a3, false, false);
    }
    #pragma unroll
    for (int r = 0; r < 8; ++r) {
      size_t m = (size_t)(mbase + r + 8 * hi);
      p.out_n[m * HDIM + lo] = sigf(a3[r]) * hv[r];   // n2*h; pass 2 kernel adds n1*neighborhood
    }
  }
}

// n = n1 * sum_c(gates*child) + (n2*h already in out_n)
__launch_bounds__(256)
__global__ void nodecell_neigh_kernel(const float4* __restrict__ gw,
                                      const float4* __restrict__ child,
                                      const float4* __restrict__ n1,
                                      float4* __restrict__ out_n,
                                      int quarter) {   // quarter = B*16/4 (float4s per plane)
  int i = blockIdx.x * blockDim.x + threadIdx.x;
  int stride = gridDim.x * blockDim.x;
  for (; i < quarter; i += stride) {
    float4 acc = make_float4(0.f, 0.f, 0.f, 0.f);
    #pragma unroll
    for (int c = 0; c < 8; ++c) {
      float4 g  = gw[(size_t)c * quarter + i];
      float4 ch = child[(size_t)c * quarter + i];
      acc.x = fmaf(g.x, ch.x, acc.x);
      acc.y = fmaf(g.y, ch.y, acc.y);
      acc.z = fmaf(g.z, ch.z, acc.z);
      acc.w = fmaf(g.w, ch.w, acc.w);
    }
    float4 a = n1[i];
    float4 b = out_n[i];
    float4 r;
    r.x = fmaf(a.x, acc.x, b.x);
    r.y = fmaf(a.y, acc.y, b.y);
    r.z = fmaf(a.z, acc.z, b.z);
    r.w = fmaf(a.w, acc.w, b.w);
    out_n[i] = r;
  }
}

extern "C" void kernel_launch(void* const* d_in, const int* in_sizes, int n_in,
                              void* d_out, int out_size, void* d_ws, size_t ws_size,
                              hipStream_t stream) {
  const int B = in_sizes[0];
  KParams p;
  p.inputs  = (const float*)d_in[0];
  p.h_prev  = (const float*)d_in[1];
  p.c_prev  = (const float*)d_in[2];
  const float* child_n = (const float*)d_in[3];
  p.w_in    = (const float*)d_in[4];   p.b_in    = (const float*)d_in[5];
  p.w_ifo_x = (const float*)d_in[6];   p.b_ifo_x = (const float*)d_in[7];
  p.w_ifo_h = (const float*)d_in[8];   p.b_ifo_h = (const float*)d_in[9];
  p.w_a_x   = (const float*)d_in[10];  p.b_a_x   = (const float*)d_in[11];
  p.w_a_h   = (const float*)d_in[12];  p.b_a_h   = (const float*)d_in[13];
  p.w_cx    = (const float*)d_in[14];  p.b_cx    = (const float*)d_in[15];
  p.w_ch    = (const float*)d_in[16];  p.b_ch    = (const float*)d_in[17];
  p.w_1x    = (const float*)d_in[18];  p.b_1x    = (const float*)d_in[19];
  p.w_1h    = (const float*)d_in[20];  p.b_1h    = (const float*)d_in[21];
  p.w_2x    = (const float*)d_in[22];  p.b_2x    = (const float*)d_in[23];
  p.w_2h    = (const float*)d_in[24];  p.b_2h    = (const float*)d_in[25];

  float* out = (float*)d_out;
  p.out_h = out;
  p.out_c = out + (size_t)B * HDIM;
  p.out_n = out + (size_t)2 * B * HDIM;

  float* ws = (float*)d_ws;
  p.gw  = ws;                           // [B,128]
  p.n1w = ws + (size_t)B * 128;         // [B,16]
  p.nTiles = B / 16;

  nodecell_gemm_kernel<<<2048, 256, 0, stream>>>(p);

  int quarter = (B * HDIM) / 4;
  nodecell_neigh_kernel<<<2048, 256, 0, stream>>>(
      (const float4*)p.gw, (const float4*)child_n, (const float4*)p.n1w,
      (float4*)p.out_n, quarter);
}